// TurboQuantValue_21801253994897
// MI455X (gfx1250) — compile-verified
//
#include <hip/hip_runtime.h>
#include <stdint.h>

// Fake-quantize (4-bit, group = row of 128 f32) + dequantize. Memory-bound:
// 256 MB in + 256 MB out -> ~22us floor at 23.3 TB/s. One wave32 handles 4
// rows in parallel (8 lanes/row); cross-lane reduction = 3 ds_swizzle XOR
// levels for all 4 rows at once. Input staged via CDNA5 async global->LDS
// copies (ASYNCcnt) with a 640B LDS row stride for bank-conflict-free
// transposed reads; outputs via nontemporal b128 stores.

typedef float v4f __attribute__((ext_vector_type(4)));

#define WAVES_PER_BLOCK 8
#define ROWS_PER_WAVE   4
#define ROW_ELTS        128
#define SLOT_V4         40     // 640 B / 16 B: 512 B row + 128 B pad (bank spread)

__device__ __forceinline__ void async_copy16(uint32_t lds_addr, const void* gptr) {
    // Per-lane 16 B global->LDS copy, tracked by ASYNCcnt. NT: input is
    // streamed exactly once, don't rinse it through L2.
    asm volatile("global_load_async_to_lds_b128 %0, %1, off th:TH_LOAD_NT"
                 :: "v"(lds_addr), "v"(gptr)
                 : "memory");
}

// ds_swizzle group-of-32 XOR mode: lane' = lane ^ M, single DS op, no index
// VALU (unlike generic __shfl_xor which materializes mbcnt+xor+bounds check).
#define SWZ_XOR(x, M) \
    __builtin_bit_cast(float, __builtin_amdgcn_ds_swizzle( \
        __builtin_bit_cast(int, (x)), ((M) << 10) | 0x1F))

__device__ __forceinline__ float min4(v4f v) {
    return fminf(fminf(v.x, v.y), fminf(v.z, v.w));
}
__device__ __forceinline__ float max4(v4f v) {
    return fmaxf(fmaxf(v.x, v.y), fmaxf(v.z, v.w));
}

__device__ __forceinline__ v4f quant4(v4f v, float lmin, float scale, float inv) {
    v4f t = (v - lmin) * inv;        // vector form -> VOPD/pk pairing
    v4f q;
    q.x = __builtin_amdgcn_fmed3f(rintf(t.x), 0.0f, 15.0f);
    q.y = __builtin_amdgcn_fmed3f(rintf(t.y), 0.0f, 15.0f);
    q.z = __builtin_amdgcn_fmed3f(rintf(t.z), 0.0f, 15.0f);
    q.w = __builtin_amdgcn_fmed3f(rintf(t.w), 0.0f, 15.0f);
    return q * scale + lmin;         // contracts to v_fma
}

__global__ __launch_bounds__(256)
void turboquant_rowq4_kernel(const float* __restrict__ x,
                             float* __restrict__ out, int rows) {
    __shared__ v4f sbuf[WAVES_PER_BLOCK][ROWS_PER_WAVE][SLOT_V4];

    const int tid  = threadIdx.x;
    const int wave = tid >> 5;
    const int lane = tid & 31;
    const long long gw   = (long long)blockIdx.x * WAVES_PER_BLOCK + wave;
    const long long row0 = gw * ROWS_PER_WAVE;
    if (row0 >= rows) return;  // wave-uniform

    if (row0 + ROWS_PER_WAVE <= (long long)rows) {
        const float* srcw = x   + row0 * ROW_ELTS;
        float*       dstw = out + row0 * ROW_ELTS;

        // Stage the whole 2 KB block (4 rows) into LDS; linear writes,
        // 640 B stride between row slots.
        const uint32_t lbase = (uint32_t)(uintptr_t)&sbuf[wave][0][0];
        const uint32_t ldst  = lbase + (uint32_t)lane * 16u;
        async_copy16(ldst,            srcw +   0 + lane * 4);
        async_copy16(ldst + 1 * 640u, srcw + 128 + lane * 4);
        async_copy16(ldst + 2 * 640u, srcw + 256 + lane * 4);
        async_copy16(ldst + 3 * 640u, srcw + 384 + lane * 4);
        asm volatile("s_wait_asynccnt 0x0" ::: "memory");

        // Transposed read: lane g*8+i owns 16 elements of row g.
        const int g = lane >> 3;
        const int i = lane & 7;
        v4f a = sbuf[wave][g][0 * 8 + i];
        v4f b = sbuf[wave][g][1 * 8 + i];
        v4f c = sbuf[wave][g][2 * 8 + i];
        v4f d = sbuf[wave][g][3 * 8 + i];

        // Lane-local reduction over 16 elements (folds into v_min3/v_max3).
        float lmin = fminf(fminf(min4(a), min4(b)), fminf(min4(c), min4(d)));
        float lmax = fmaxf(fmaxf(max4(a), max4(b)), fmaxf(max4(c), max4(d)));

        // 3-level XOR butterfly within each 8-lane row group (all 4 rows).
        lmin = fminf(lmin, SWZ_XOR(lmin, 1));
        lmax = fmaxf(lmax, SWZ_XOR(lmax, 1));
        lmin = fminf(lmin, SWZ_XOR(lmin, 2));
        lmax = fmaxf(lmax, SWZ_XOR(lmax, 2));
        lmin = fminf(lmin, SWZ_XOR(lmin, 4));
        lmax = fmaxf(lmax, SWZ_XOR(lmax, 4));

        const float scale = (lmax - lmin) * (1.0f / 15.0f);
        const float inv   = __builtin_amdgcn_rcpf(scale + 1e-10f);

        // Store back: row g, chunk k, sub-pos i -> v4f index g*32 + k*8 + i.
        v4f* dq = (v4f*)dstw;
        __builtin_nontemporal_store(quant4(a, lmin, scale, inv), dq + g * 32 + 0 * 8 + i);
        __builtin_nontemporal_store(quant4(b, lmin, scale, inv), dq + g * 32 + 1 * 8 + i);
        __builtin_nontemporal_store(quant4(c, lmin, scale, inv), dq + g * 32 + 2 * 8 + i);
        __builtin_nontemporal_store(quant4(d, lmin, scale, inv), dq + g * 32 + 3 * 8 + i);
    } else {
        // Tail (absent for the reference shape): one row across 32 lanes,
        // direct loads, 5-level butterfly.
        for (long long r = row0; r < rows; ++r) {
            v4f v = ((const v4f*)(x + r * ROW_ELTS))[lane];
            float lmin = min4(v);
            float lmax = max4(v);
            lmin = fminf(lmin, SWZ_XOR(lmin, 1));
            lmax = fmaxf(lmax, SWZ_XOR(lmax, 1));
            lmin = fminf(lmin, SWZ_XOR(lmin, 2));
            lmax = fmaxf(lmax, SWZ_XOR(lmax, 2));
            lmin = fminf(lmin, SWZ_XOR(lmin, 4));
            lmax = fmaxf(lmax, SWZ_XOR(lmax, 4));
            lmin = fminf(lmin, SWZ_XOR(lmin, 8));
            lmax = fmaxf(lmax, SWZ_XOR(lmax, 8));
            lmin = fminf(lmin, SWZ_XOR(lmin, 16));
            lmax = fmaxf(lmax, SWZ_XOR(lmax, 16));
            const float scale = (lmax - lmin) * (1.0f / 15.0f);
            const float inv   = __builtin_amdgcn_rcpf(scale + 1e-10f);
            __builtin_nontemporal_store(quant4(v, lmin, scale, inv),
                                        (v4f*)(out + r * ROW_ELTS) + lane);
        }
    }
}

extern "C" void kernel_launch(void* const* d_in, const int* in_sizes, int n_in,
                              void* d_out, int out_size, void* d_ws, size_t ws_size,
                              hipStream_t stream) {
    const float* x  = (const float*)d_in[0];
    float* out      = (float*)d_out;
    const long long n = in_sizes[0];
    const int rows  = (int)(n / ROW_ELTS);

    const int waves  = (rows + ROWS_PER_WAVE - 1) / ROWS_PER_WAVE;
    const int blocks = (waves + WAVES_PER_BLOCK - 1) / WAVES_PER_BLOCK;

    turboquant_rowq4_kernel<<<blocks, 256, 0, stream>>>(x, out, rows);
}